// MuellerMatrixPyramid_23802708754564
// MI455X (gfx1250) — compile-verified
//
#include <hip/hip_runtime.h>
#include <hip/hip_bf16.h>

typedef float v2f __attribute__((ext_vector_type(2)));
typedef float v8f __attribute__((ext_vector_type(8)));

#define WAVE_SZ 32
#define WPB 4                       // waves per block for the mm kernel
#define MM_BLK (WAVE_SZ * WPB)      // 128 threads
#define PXSTR 17                    // padded LDS row per pixel (17 coprime 64 -> conflict-free)
#define BUF_FLOATS (32 * PXSTR)     // one per-wave buffer: 32 pixels x 17

// ---------------------------------------------------------------------------
// 4x4 inverse via adjugate (row-major). Matrices are diagonally dominant per
// setup_inputs(), so this is numerically safe.
// ---------------------------------------------------------------------------
__device__ __forceinline__ void inv4x4(const float m[16], float inv[16]) {
  inv[0]  =  m[5]*m[10]*m[15] - m[5]*m[11]*m[14] - m[9]*m[6]*m[15] + m[9]*m[7]*m[14] + m[13]*m[6]*m[11] - m[13]*m[7]*m[10];
  inv[4]  = -m[4]*m[10]*m[15] + m[4]*m[11]*m[14] + m[8]*m[6]*m[15] - m[8]*m[7]*m[14] - m[12]*m[6]*m[11] + m[12]*m[7]*m[10];
  inv[8]  =  m[4]*m[9]*m[15]  - m[4]*m[11]*m[13] - m[8]*m[5]*m[15] + m[8]*m[7]*m[13] + m[12]*m[5]*m[11] - m[12]*m[7]*m[9];
  inv[12] = -m[4]*m[9]*m[14]  + m[4]*m[10]*m[13] + m[8]*m[5]*m[14] - m[8]*m[6]*m[13] - m[12]*m[5]*m[10] + m[12]*m[6]*m[9];
  inv[1]  = -m[1]*m[10]*m[15] + m[1]*m[11]*m[14] + m[9]*m[2]*m[15] - m[9]*m[3]*m[14] - m[13]*m[2]*m[11] + m[13]*m[3]*m[10];
  inv[5]  =  m[0]*m[10]*m[15] - m[0]*m[11]*m[14] - m[8]*m[2]*m[15] + m[8]*m[3]*m[14] + m[12]*m[2]*m[11] - m[12]*m[3]*m[10];
  inv[9]  = -m[0]*m[9]*m[15]  + m[0]*m[11]*m[13] + m[8]*m[1]*m[15] - m[8]*m[3]*m[13] - m[12]*m[1]*m[11] + m[12]*m[3]*m[9];
  inv[13] =  m[0]*m[9]*m[14]  - m[0]*m[10]*m[13] - m[8]*m[1]*m[14] + m[8]*m[2]*m[13] + m[12]*m[1]*m[10] - m[12]*m[2]*m[9];
  inv[2]  =  m[1]*m[6]*m[15]  - m[1]*m[7]*m[14]  - m[5]*m[2]*m[15] + m[5]*m[3]*m[14] + m[13]*m[2]*m[7]  - m[13]*m[3]*m[6];
  inv[6]  = -m[0]*m[6]*m[15]  + m[0]*m[7]*m[14]  + m[4]*m[2]*m[15] - m[4]*m[3]*m[14] - m[12]*m[2]*m[7]  + m[12]*m[3]*m[6];
  inv[10] =  m[0]*m[5]*m[15]  - m[0]*m[7]*m[13]  - m[4]*m[1]*m[15] + m[4]*m[3]*m[13] + m[12]*m[1]*m[7]  - m[12]*m[3]*m[5];
  inv[14] = -m[0]*m[5]*m[14]  + m[0]*m[6]*m[13]  + m[4]*m[1]*m[14] - m[4]*m[2]*m[13] - m[12]*m[1]*m[6]  + m[12]*m[2]*m[5];
  inv[3]  = -m[1]*m[6]*m[11]  + m[1]*m[7]*m[10]  + m[5]*m[2]*m[11] - m[5]*m[3]*m[10] - m[9]*m[2]*m[7]   + m[9]*m[3]*m[6];
  inv[7]  =  m[0]*m[6]*m[11]  - m[0]*m[7]*m[10]  - m[4]*m[2]*m[11] + m[4]*m[3]*m[10] + m[8]*m[2]*m[7]   - m[8]*m[3]*m[6];
  inv[11] = -m[0]*m[5]*m[11]  + m[0]*m[7]*m[9]   + m[4]*m[1]*m[11] - m[4]*m[3]*m[9]  - m[8]*m[1]*m[7]   + m[8]*m[3]*m[5];
  inv[15] =  m[0]*m[5]*m[10]  - m[0]*m[6]*m[9]   - m[4]*m[1]*m[10] + m[4]*m[2]*m[9]  + m[8]*m[1]*m[6]   - m[8]*m[2]*m[5];
  float det = m[0]*inv[0] + m[1]*inv[4] + m[2]*inv[8] + m[3]*inv[12];
  float r = 1.0f / det;
#pragma unroll
  for (int i = 0; i < 16; ++i) inv[i] *= r;
}

// ---------------------------------------------------------------------------
// Mueller features: per pixel, M = inv(A) * I * inv(W), normalized by M00.
// Each wave owns 32 pixels. Inverses are per-lane VALU; the two 4x4 matmul
// chains run on V_WMMA_F32_16X16X4_F32, 4 pixels per 16x16 tile (block-
// diagonal packing), 8 tiles per stage per wave. Lane<->matrix marshaling
// goes through per-wave LDS buffers.
//
// WMMA f32 16x16x4 lane layouts (ISA 7.12.2):
//   A(16x4):  VGPR v, lane L:      A[M = L&15][K = 2*(L>>4) + v]
//   B(4x16):  VGPR v, lane L:      B[K = 2*(L>>4) + v][N = L&15]
//   D(16x16): VGPR r, lane L:      D[M = r + 8*(L>>4)][N = L&15]
// ---------------------------------------------------------------------------
__global__ void __launch_bounds__(MM_BLK)
mm_features_kernel(const float* __restrict__ x, float* __restrict__ out,
                   int hw, int outC) {
  __shared__ float smem[WPB * 4 * BUF_FLOATS];   // 34,816 B
  const int lane = threadIdx.x & 31;
  const int wid  = threadIdx.x >> 5;
  float* bufI = smem + wid * (4 * BUF_FLOATS);
  float* bufA = bufI + BUF_FLOATS;   // inv(A)
  float* bufW = bufA + BUF_FLOATS;   // inv(W)
  float* bufT = bufW + BUF_FLOATS;   // T = inv(A)*I
  float* bufM = bufA;                // alias: inv(A) dead after stage 1

  const int pix = blockIdx.x * MM_BLK + threadIdx.x;   // grids divide exactly
  const int b   = pix / hw;
  const int rem = pix - b * hw;
  const float* px = x + (size_t)b * 48 * (size_t)hw + rem;

  float I[16], A[16], Wm[16];
#pragma unroll
  for (int e = 0; e < 16; ++e) {
    I[e]  = px[(size_t)e * hw];
    Wm[e] = px[(size_t)(32 + e) * hw];
    A[e]  = px[(size_t)(16 + e) * hw];
  }
  float inten = 0.f;
#pragma unroll
  for (int e = 0; e < 16; ++e) inten += I[e];
  inten *= (1.0f / 16.0f);

  float iA[16], iW[16];
  inv4x4(A, iA);
  inv4x4(Wm, iW);

#pragma unroll
  for (int e = 0; e < 16; ++e) {
    bufI[lane * PXSTR + e] = I[e];
    bufA[lane * PXSTR + e] = iA[e];
    bufW[lane * PXSTR + e] = iW[e];
  }
  __syncthreads();

  // operand-gather indices (per ISA layout above)
  const int mrow  = lane & 15;            // M for A-op, N for B-op
  const int khalf = (lane >> 4) << 1;     // K base: 0 or 2
  const int prow  = mrow >> 2;            // pixel within tile (0..3)
  const int i4    = (mrow & 3) * 4;       // row*4 inside a 4x4 block
  const int j4    = (mrow & 3);           // column inside a 4x4 block
  // diagonal-block extraction roles (lanes 0-7 and 24-31 carry useful D data)
  const bool act = (lane < 8) || (lane >= 24);
  const int  pl  = ((lane & 16) >> 3) | ((lane >> 2) & 1);  // pixel 0..3 in tile
  const int  jj  = lane & 3;                                 // column j
  // lanes with (lane>>2)&1 == 1 take D rows from d[4..7]; bitwise select mask
  // (value-level select; avoids InstCombine folding into a dynamic vector
  //  extract which lowers as a huge v_cmp/v_cndmask cascade)
  const unsigned selmask = 0u - (((unsigned)lane >> 2) & 1u);

  const v8f cz = {0.f, 0.f, 0.f, 0.f, 0.f, 0.f, 0.f, 0.f};

  // ---- stage 1: T_p = inv(A_p) * I_p ----
#pragma unroll
  for (int t = 0; t < 8; ++t) {
    const int base = t * 4;
    v2f a, bb;
    a[0]  = bufA[(base + prow) * PXSTR + i4 + khalf];
    a[1]  = bufA[(base + prow) * PXSTR + i4 + khalf + 1];
    bb[0] = bufI[(base + prow) * PXSTR + khalf * 4 + j4];
    bb[1] = bufI[(base + prow) * PXSTR + (khalf + 1) * 4 + j4];
    v8f d = __builtin_amdgcn_wmma_f32_16x16x4_f32(false, a, false, bb,
                                                  (short)0, cz, false, false);
    if (act) {
      const int pg = (base + pl) * PXSTR + jj;
#pragma unroll
      for (int i = 0; i < 4; ++i) {
        const unsigned lo = __float_as_uint(d[i]);
        const unsigned hv = __float_as_uint(d[4 + i]);
        bufT[pg + i * 4] = __uint_as_float((hv & selmask) | (lo & ~selmask));
      }
    }
  }
  __syncthreads();

  // ---- stage 2: M_p = T_p * inv(W_p) ----
#pragma unroll
  for (int t = 0; t < 8; ++t) {
    const int base = t * 4;
    v2f a, bb;
    a[0]  = bufT[(base + prow) * PXSTR + i4 + khalf];
    a[1]  = bufT[(base + prow) * PXSTR + i4 + khalf + 1];
    bb[0] = bufW[(base + prow) * PXSTR + khalf * 4 + j4];
    bb[1] = bufW[(base + prow) * PXSTR + (khalf + 1) * 4 + j4];
    v8f d = __builtin_amdgcn_wmma_f32_16x16x4_f32(false, a, false, bb,
                                                  (short)0, cz, false, false);
    if (act) {
      const int pg = (base + pl) * PXSTR + jj;
#pragma unroll
      for (int i = 0; i < 4; ++i) {
        const unsigned lo = __float_as_uint(d[i]);
        const unsigned hv = __float_as_uint(d[4 + i]);
        bufM[pg + i * 4] = __uint_as_float((hv & selmask) | (lo & ~selmask));
      }
    }
  }
  __syncthreads();

  // ---- normalize + store 17 channels ----
  const float rn = 1.0f / bufM[lane * PXSTR];
  float* o = out + (size_t)b * outC * (size_t)hw + rem;
  o[0] = inten;
#pragma unroll
  for (int e = 0; e < 16; ++e)
    o[(size_t)(1 + e) * hw] = bufM[lane * PXSTR + e] * rn;
}

// ---------------------------------------------------------------------------
// 4x4 max-pool, stride 4, VALID (pyramid downsample). in: (BC, h, w).
// ---------------------------------------------------------------------------
__global__ void maxpool4_kernel(const float* __restrict__ in, float* __restrict__ out,
                                int h, int w, long long total) {
  long long idx = (long long)blockIdx.x * blockDim.x + threadIdx.x;
  if (idx >= total) return;
  const int oh = h >> 2, ow = w >> 2;
  const int oxy = (int)(idx % (long long)(oh * ow));
  const long long bc = idx / (long long)(oh * ow);
  const int oy = oxy / ow, ox = oxy - oy * ow;
  const float* p = in + bc * (size_t)h * w + (size_t)(oy * 4) * w + (ox * 4);
  float mx = -__builtin_inff();
#pragma unroll
  for (int dy = 0; dy < 4; ++dy)
#pragma unroll
    for (int dx = 0; dx < 4; ++dx)
      mx = fmaxf(mx, p[(size_t)dy * w + dx]);
  out[idx] = mx;
}

// ---------------------------------------------------------------------------
// Bilinear upsample, align_corners=True. src: (B,17,h,w) -> out channels
// [c0, c0+17) of (B,51,H,W).
// ---------------------------------------------------------------------------
__global__ void upsample_ac_kernel(const float* __restrict__ src, float* __restrict__ out,
                                   int h, int w, int H, int W,
                                   int outC, int c0, long long total) {
  long long idx = (long long)blockIdx.x * blockDim.x + threadIdx.x;
  if (idx >= total) return;
  const int X = (int)(idx % W);
  long long r = idx / W;
  const int Y = (int)(r % H);  r /= H;
  const int c = (int)(r % 17);
  const int b = (int)(r / 17);

  const float sy = (float)Y * (float)(h - 1) / (float)(H - 1);
  const float sx = (float)X * (float)(w - 1) / (float)(W - 1);
  int y0 = (int)sy;  if (y0 > h - 1) y0 = h - 1;
  int x0 = (int)sx;  if (x0 > w - 1) x0 = w - 1;
  const int y1 = (y0 + 1 < h) ? y0 + 1 : h - 1;
  const int x1 = (x0 + 1 < w) ? x0 + 1 : w - 1;
  const float fy = sy - (float)y0;
  const float fx = sx - (float)x0;

  const float* s = src + ((size_t)b * 17 + c) * (size_t)h * w;
  const float v00 = s[(size_t)y0 * w + x0];
  const float v01 = s[(size_t)y0 * w + x1];
  const float v10 = s[(size_t)y1 * w + x0];
  const float v11 = s[(size_t)y1 * w + x1];
  const float top = v00 + (v01 - v00) * fx;
  const float bot = v10 + (v11 - v10) * fx;
  out[((size_t)b * outC + c0 + c) * (size_t)H * W + (size_t)Y * W + X] =
      top + (bot - top) * fy;
}

// ---------------------------------------------------------------------------
extern "C" void kernel_launch(void* const* d_in, const int* in_sizes, int n_in,
                              void* d_out, int out_size, void* d_ws, size_t ws_size,
                              hipStream_t stream) {
  const float* x = (const float*)d_in[0];
  float* out = (float*)d_out;

  const int B = 4, C = 48, H = 512, W = 512;
  const int OUTC = 51;  // 17 * 3 levels

  // workspace layout (floats): ~18.1 MB total
  float* cur1 = (float*)d_ws;                                  // (4,48,128,128)
  float* cur2 = cur1 + (size_t)B * C * 128 * 128;              // (4,48,32,32)
  float* m1   = cur2 + (size_t)B * C * 32 * 32;                // (4,17,128,128)
  float* m2   = m1   + (size_t)B * 17 * 128 * 128;             // (4,17,32,32)

  // --- level 0: features straight into d_out channels [0,17) (h==H) ---
  {
    const long long N = (long long)B * H * W;                  // 1,048,576
    mm_features_kernel<<<(int)(N / MM_BLK), MM_BLK, 0, stream>>>(x, out, H * W, OUTC);
  }
  // --- pool 512 -> 128 ---
  {
    const long long N = (long long)B * C * 128 * 128;
    maxpool4_kernel<<<(int)((N + 255) / 256), 256, 0, stream>>>(x, cur1, 512, 512, N);
  }
  // --- level 1: features at 128, upsample into channels [17,34) ---
  {
    const long long N = (long long)B * 128 * 128;
    mm_features_kernel<<<(int)(N / MM_BLK), MM_BLK, 0, stream>>>(cur1, m1, 128 * 128, 17);
    const long long M = (long long)B * 17 * H * W;
    upsample_ac_kernel<<<(int)((M + 255) / 256), 256, 0, stream>>>(m1, out, 128, 128, H, W, OUTC, 17, M);
  }
  // --- pool 128 -> 32 ---
  {
    const long long N = (long long)B * C * 32 * 32;
    maxpool4_kernel<<<(int)((N + 255) / 256), 256, 0, stream>>>(cur1, cur2, 128, 128, N);
  }
  // --- level 2: features at 32, upsample into channels [34,51) ---
  {
    const long long N = (long long)B * 32 * 32;
    mm_features_kernel<<<(int)(N / MM_BLK), MM_BLK, 0, stream>>>(cur2, m2, 32 * 32, 17);
    const long long M = (long long)B * 17 * H * W;
    upsample_ac_kernel<<<(int)((M + 255) / 256), 256, 0, stream>>>(m2, out, 32, 32, H, W, OUTC, 34, M);
  }
}